// AnomalyAttention_72284299591901
// MI455X (gfx1250) — compile-verified
//
#include <hip/hip_runtime.h>
#include <cmath>

typedef __attribute__((ext_vector_type(2))) float v2f;
typedef __attribute__((ext_vector_type(8))) float v8f;

#define BN 4
#define LN 1024
#define HN 8
#define EN 64

__device__ __forceinline__ v8f wmma_f32(v2f a, v2f b, v8f c) {
  // V_WMMA_F32_16X16X4_F32: D = A(16x4) * B(4x16) + C(16x16), all fp32
  return __builtin_amdgcn_wmma_f32_16x16x4_f32(false, a, false, b, (short)0, c,
                                               false, false);
}

// ---------------------------------------------------------------------------
// Kernel 1: sig = 3^(sigmoid(5*sigma)+1e-5) - 1   (output, [B,H,L])
//           pd  = inv_sqrt(2pi)/sig * exp(-x^2/(2 sig^2)) + 1e-5  (-> ws)
// pd layout: [bh][l][e] contiguous. One thread per element.
// ---------------------------------------------------------------------------
__global__ __launch_bounds__(256) void sig_pd_kernel(
    const float* __restrict__ sigma, const float* __restrict__ x,
    float* __restrict__ pd, float* __restrict__ sig_out) {
  const int idx = blockIdx.x * blockDim.x + threadIdx.x;  // B*H*L*E threads
  const int e = idx & (EN - 1);
  const int bhl = idx >> 6;            // (b*H+h)*L + l
  const int bh = bhl >> 10;            // L = 1024
  const int l = bhl & (LN - 1);
  const int b = bh >> 3;               // H = 8
  const int h = bh & (HN - 1);

  float sg = sigma[(b * LN + l) * HN + h];
  sg = 1.0f / (1.0f + __expf(-5.0f * sg)) + 1e-5f;   // sigmoid(5x)+1e-5
  sg = __expf(sg * 1.0986122886681098f) - 1.0f;      // 3^sg - 1

  const float xv = x[((size_t)(b * LN + l) * HN + h) * EN + e];
  const float pdv =
      0.3989422804014327f / sg * __expf(-(xv * xv) / (2.0f * sg * sg)) + 1e-5f;
  pd[idx] = pdv;
  if (e == 0) sig_out[bhl] = sg;
}

// ---------------------------------------------------------------------------
// Kernel 2: causal attention. One wave (32 threads) per (bh, 16-row strip).
// Scores via WMMA f32 -> LDS (16x1024), softmax with shfl reductions,
// write series (coalesced, zero-fill masked region), then V = P * values.
// ---------------------------------------------------------------------------
__global__ __launch_bounds__(32) void attn_kernel(
    const float* __restrict__ q, const float* __restrict__ kk,
    const float* __restrict__ vv, float* __restrict__ outV,
    float* __restrict__ outSeries) {
  __shared__ float S[16][LN];  // 64 KB score strip

  const int mtile = blockIdx.x & 63;
  const int bh = blockIdx.x >> 6;
  const int b = bh >> 3, h = bh & 7;
  const int lane = threadIdx.x;
  const int half = lane >> 4;   // which 16-lane half
  const int r16 = lane & 15;
  const float scale = 0.125f;   // 1/sqrt(64)

  // A fragment: Q rows of this strip. Lane holds Q[m=r16][4g+2*half +{0,1}].
  const int row_m = mtile * 16 + r16;
  const float* Qrow = q + ((size_t)(b * LN + row_m) * HN + h) * EN;
  v2f aQ[16];
#pragma unroll
  for (int g = 0; g < 16; ++g)
    aQ[g] = *(const v2f*)(Qrow + 4 * g + 2 * half);

  const int ntiles = mtile + 1;  // causal: only tiles up to the diagonal
  for (int ct = 0; ct < ntiles; ++ct) {
    // B fragment: K rows (Kᵀ columns). Same per-lane pattern as A.
    const float* Krow = kk + ((size_t)(b * LN + ct * 16 + r16) * HN + h) * EN;
    v8f acc = {};
#pragma unroll
    for (int g = 0; g < 16; ++g) {
      const v2f bK = *(const v2f*)(Krow + 4 * g + 2 * half);
      acc = wmma_f32(aQ[g], bK, acc);
    }
    // D layout: row = v + 8*half, col = r16. Scale, causal-mask, stash in LDS.
#pragma unroll
    for (int v = 0; v < 8; ++v) {
      const int row = v + 8 * half;
      float s = acc[v] * scale;
      if (ct == mtile && r16 > row) s = -INFINITY;
      S[row][ct * 16 + r16] = s;
    }
  }
  __syncthreads();

  // Softmax stats: lane pair (r16, r16+16) shares row r16, strided columns.
  const int ncols = ntiles * 16;
  float m = -INFINITY;
  for (int c = half; c < ncols; c += 2) m = fmaxf(m, S[r16][c]);
  m = fmaxf(m, __shfl_xor(m, 16));
  float ssum = 0.0f;
  for (int c = half; c < ncols; c += 2) ssum += __expf(S[r16][c] - m);
  ssum += __shfl_xor(ssum, 16);
  const float inv = 1.0f / ssum;

  // Normalize + write series row-major (coalesced 128B per step per row).
#pragma unroll 1
  for (int r = 0; r < 16; ++r) {
    const float m_r = __shfl(m, r);
    const float inv_r = __shfl(inv, r);
    float* srow = outSeries + ((size_t)bh * LN + mtile * 16 + r) * LN;
    for (int c = lane; c < ncols; c += 32) {
      const float p = __expf(S[r][c] - m_r) * inv_r;
      S[r][c] = p;   // keep probabilities for the V matmul
      srow[c] = p;
    }
    for (int c = ncols + lane; c < LN; c += 32) srow[c] = 0.0f;  // masked region
  }
  __syncthreads();

  // V = P(16xL) * values(Lx64): four 16x16 D accumulators, K in steps of 4.
  v8f accV[4] = {{}, {}, {}, {}};
  for (int ct = 0; ct < ntiles; ++ct) {
    v2f aP[4];
#pragma unroll
    for (int g = 0; g < 4; ++g)
      aP[g] = *(const v2f*)(&S[r16][ct * 16 + 4 * g + 2 * half]);
    const float* Vb = vv + ((size_t)(b * LN + ct * 16) * HN + h) * EN;
#pragma unroll
    for (int c = 0; c < 4; ++c) {
#pragma unroll
      for (int g = 0; g < 4; ++g) {
        v2f bV;
        bV.x = Vb[(size_t)(4 * g + 2 * half) * (HN * EN) + c * 16 + r16];
        bV.y = Vb[(size_t)(4 * g + 2 * half + 1) * (HN * EN) + c * 16 + r16];
        accV[c] = wmma_f32(aP[g], bV, accV[c]);
      }
    }
  }
  // Write V: [B, L, H, 64], row = v + 8*half, col d = c*16 + r16 (coalesced).
#pragma unroll
  for (int c = 0; c < 4; ++c)
#pragma unroll
    for (int v = 0; v < 8; ++v) {
      const int row = mtile * 16 + v + 8 * half;
      outV[((size_t)(b * LN + row) * HN + h) * EN + c * 16 + r16] = accV[c][v];
    }
}

// ---------------------------------------------------------------------------
// Kernel 3: prior = pd * pdᵀ per (b,h). 4 waves per 16-row strip; each wave
// keeps its A fragment resident and sweeps every 4th column tile.
// ---------------------------------------------------------------------------
__global__ __launch_bounds__(128) void prior_kernel(
    const float* __restrict__ pd, float* __restrict__ outPrior) {
  const int mtile = blockIdx.x & 63;
  const int bh = blockIdx.x >> 6;
  const int tid = threadIdx.x;
  const int wave = tid >> 5;
  const int lane = tid & 31;
  const int half = lane >> 4;
  const int r16 = lane & 15;

  const float* Ar = pd + ((size_t)bh * LN + mtile * 16 + r16) * EN;
  v2f aA[16];
#pragma unroll
  for (int g = 0; g < 16; ++g)
    aA[g] = *(const v2f*)(Ar + 4 * g + 2 * half);

  for (int ct = wave; ct < 64; ct += 4) {
    const float* Br = pd + ((size_t)bh * LN + ct * 16 + r16) * EN;
    v8f acc = {};
#pragma unroll
    for (int g = 0; g < 16; ++g) {
      const v2f bB = *(const v2f*)(Br + 4 * g + 2 * half);
      acc = wmma_f32(aA[g], bB, acc);
    }
#pragma unroll
    for (int v = 0; v < 8; ++v) {
      const int row = mtile * 16 + v + 8 * half;
      outPrior[((size_t)bh * LN + row) * LN + ct * 16 + r16] = acc[v];
    }
  }
}

// ---------------------------------------------------------------------------
extern "C" void kernel_launch(void* const* d_in, const int* in_sizes, int n_in,
                              void* d_out, int out_size, void* d_ws,
                              size_t ws_size, hipStream_t stream) {
  (void)in_sizes; (void)n_in; (void)out_size; (void)ws_size;
  const float* q     = (const float*)d_in[0];
  const float* k     = (const float*)d_in[1];
  const float* v     = (const float*)d_in[2];
  const float* sigma = (const float*)d_in[3];
  const float* x     = (const float*)d_in[4];
  // d_in[5] = k (top-k count) — unused by the reference computation.

  float* out = (float*)d_out;
  float* outV      = out;                                        // [B,L,H,64]
  float* outSeries = outV + (size_t)BN * LN * HN * EN;           // [B,H,L,L]
  float* outPrior  = outSeries + (size_t)BN * HN * LN * LN;      // [B,H,L,L]
  float* outSig    = outPrior + (size_t)BN * HN * LN * LN;       // [B,H,L,1]

  float* pd = (float*)d_ws;  // [B*H][L][64] = 8 MB scratch

  sig_pd_kernel<<<(BN * HN * LN * EN) / 256, 256, 0, stream>>>(sigma, x, pd,
                                                               outSig);
  attn_kernel<<<BN * HN * 64, 32, 0, stream>>>(q, k, v, outV, outSeries);
  prior_kernel<<<BN * HN * 64, 128, 0, stream>>>(pd, outPrior);
}